// GCNN_63462436766296
// MI455X (gfx1250) — compile-verified
//
#include <hip/hip_runtime.h>

#define D 128
#define NEG_SLOPE 0.2f

typedef __attribute__((ext_vector_type(2))) float v2f;
typedef __attribute__((ext_vector_type(8))) float v8f;

// ---- ordered-uint encoding for exact float atomic max ----
__device__ __forceinline__ unsigned f2ord(float f) {
    unsigned u = __float_as_uint(f);
    return (u & 0x80000000u) ? ~u : (u | 0x80000000u);
}
__device__ __forceinline__ float ord2f(unsigned u) {
    return (u & 0x80000000u) ? __uint_as_float(u & 0x7fffffffu)
                             : __uint_as_float(~u);
}

// ================= GEMM: hw = h @ w  (fp32 WMMA 16x16x4) =================
// grid.x = ceil(n/16); block = 256 (8 waves). Wave i computes cols [16i,16i+16)
// of one 16-row strip. A: 16x4 f32 (2 VGPR), B: 4x16 f32 (2 VGPR), C: 16x16 f32.
// Out-of-range A rows are only address-clamped (not zeroed): C-row M depends
// solely on A-row M, and the store epilogue masks rows >= n, so garbage rows
// never reach memory. This keeps the load->wmma chain free of cndmasks.
__global__ void gat_gemm_wmma(const float* __restrict__ h,
                              const float* __restrict__ w,
                              float* __restrict__ hw, int n) {
    const int wave = threadIdx.x >> 5;        // 0..7 -> column tile
    const int lane = threadIdx.x & 31;
    const int m    = lane & 15;               // row-in-tile (A) / col-in-tile (B,C)
    const int kh   = (lane >> 4) * 2;         // 0 or 2: K sub-offset per lane half
    const int row0 = blockIdx.x * 16;
    const int col0 = wave * 16;

    const int arow = row0 + m;
    const float* __restrict__ ap = h + (size_t)(arow < n ? arow : 0) * D;

    v8f c = {};
    #pragma unroll
    for (int k0 = 0; k0 < D; k0 += 4) {
        const int k = k0 + kh;
        v2f a, b;
        a[0] = ap[k];
        a[1] = ap[k + 1];
        b[0] = w[(size_t)k * D + col0 + m];
        b[1] = w[(size_t)(k + 1) * D + col0 + m];
        c = __builtin_amdgcn_wmma_f32_16x16x4_f32(false, a, false, b,
                                                  (short)0, c, false, false);
    }
    // C layout: VGPR r, lanes 0-15 -> M=r, lanes 16-31 -> M=r+8; N = lane&15
    const int rbase = row0 + 8 * (lane >> 4);
    const int cc = col0 + m;
    #pragma unroll
    for (int r = 0; r < 8; ++r) {
        const int orow = rbase + r;
        if (orow < n) hw[(size_t)orow * D + cc] = c[r];
    }
}

// ======== per-node attention logits: esrc = hw.a_src, edst = hw.a_dst =====
// one wave per node; lane reads float4, shuffle-reduce.
__global__ void gat_node_logits(const float* __restrict__ hw,
                                const float* __restrict__ a_src,
                                const float* __restrict__ a_dst,
                                float* __restrict__ esrc,
                                float* __restrict__ edst, int n) {
    const int node = (blockIdx.x * blockDim.x + threadIdx.x) >> 5;
    const int lane = threadIdx.x & 31;
    if (node >= n) return;
    const float4 v  = ((const float4*)(hw + (size_t)node * D))[lane];
    const float4 as = ((const float4*)a_src)[lane];
    const float4 ad = ((const float4*)a_dst)[lane];
    float s = v.x * as.x + v.y * as.y + v.z * as.z + v.w * as.w;
    float d = v.x * ad.x + v.y * ad.y + v.z * ad.z + v.w * ad.w;
    #pragma unroll
    for (int off = 16; off > 0; off >>= 1) {
        s += __shfl_down(s, off);
        d += __shfl_down(d, off);
    }
    if (lane == 0) { esrc[node] = s; edst[node] = d; }
}

// ======== pass B: edge logit = leaky_relu(esrc[s]+edst[d]); segment max ====
__global__ void gat_edge_logit_max(const int* __restrict__ ei, int E, int etot,
                                   const float* __restrict__ esrc,
                                   const float* __restrict__ edst,
                                   float* __restrict__ logit,
                                   unsigned* __restrict__ mord) {
    const int e = blockIdx.x * blockDim.x + threadIdx.x;
    if (e >= etot) return;
    int s, d;
    if (e < E) { s = ei[e]; d = ei[E + e]; }
    else       { s = e - E; d = s; }              // self-loop
    float v = esrc[s] + edst[d];
    v = (v > 0.f) ? v : NEG_SLOPE * v;
    logit[e] = v;
    atomicMax(mord + d, f2ord(v));
}

// ======== pass C: p = exp(logit - m[dst]); segment sum into denom =========
__global__ void gat_edge_exp_sum(const int* __restrict__ ei, int E, int etot,
                                 const unsigned* __restrict__ mord,
                                 float* __restrict__ pbuf,   // in: logit, out: p
                                 float* __restrict__ denom) {
    const int e = blockIdx.x * blockDim.x + threadIdx.x;
    if (e >= etot) return;
    const int d = (e < E) ? ei[E + e] : (e - E);
    const float p = __expf(pbuf[e] - ord2f(mord[d]));
    pbuf[e] = p;
    unsafeAtomicAdd(denom + d, p);
}

// ======== pass D: out[dst] += alpha * hw[src]  (wave per edge) ============
__global__ void gat_edge_scatter(const int* __restrict__ ei, int E, int etot,
                                 const float* __restrict__ pbuf,
                                 const float* __restrict__ denom,
                                 const float* __restrict__ hw,
                                 float* __restrict__ out) {
    const int e    = (blockIdx.x * blockDim.x + threadIdx.x) >> 5;
    const int lane = threadIdx.x & 31;
    if (e >= etot) return;
    int s, d;
    if (e < E) { s = ei[e]; d = ei[E + e]; }
    else       { s = e - E; d = s; }
    const float alpha = pbuf[e] / denom[d];
    const float4 hv = ((const float4*)(hw + (size_t)s * D))[lane];
    float* o = out + (size_t)d * D + lane * 4;
    unsafeAtomicAdd(o + 0, hv.x * alpha);
    unsafeAtomicAdd(o + 1, hv.y * alpha);
    unsafeAtomicAdd(o + 2, hv.z * alpha);
    unsafeAtomicAdd(o + 3, hv.w * alpha);
}

// ======== bias + relu =====================================================
__global__ void gat_bias_relu(float* __restrict__ h, const float* __restrict__ b,
                              int total) {
    const int i = blockIdx.x * blockDim.x + threadIdx.x;
    if (i >= total) return;
    const float v = h[i] + b[i & (D - 1)];
    h[i] = v > 0.f ? v : 0.f;
}

// ======== mean pool =======================================================
__global__ void gat_colsum(const float* __restrict__ h, float* __restrict__ out,
                           int n) {
    const int col  = threadIdx.x & (D - 1);
    const int half = threadIdx.x >> 7;        // blockDim = 256 -> 2 row groups
    float s = 0.f;
    for (int r = blockIdx.x * 2 + half; r < n; r += gridDim.x * 2)
        s += h[(size_t)r * D + col];
    unsafeAtomicAdd(out + col, s);
}
__global__ void gat_scale(float* out, float inv) { out[threadIdx.x] *= inv; }

// ==========================================================================
extern "C" void kernel_launch(void* const* d_in, const int* in_sizes, int n_in,
                              void* d_out, int out_size, void* d_ws, size_t ws_size,
                              hipStream_t stream) {
    const float* x  = (const float*)d_in[0];
    const int*   ei = (const int*)d_in[1];
    const float* w[3]    = {(const float*)d_in[2], (const float*)d_in[6],  (const float*)d_in[10]};
    const float* asrc[3] = {(const float*)d_in[3], (const float*)d_in[7],  (const float*)d_in[11]};
    const float* adst[3] = {(const float*)d_in[4], (const float*)d_in[8],  (const float*)d_in[12]};
    const float* bias[3] = {(const float*)d_in[5], (const float*)d_in[9],  (const float*)d_in[13]};

    const int n    = in_sizes[0] / D;
    const int E    = in_sizes[1] / 2;
    const int etot = E + n;

    // workspace layout
    float*    bufA  = (float*)d_ws;                    // hw          [n*D]
    float*    bufB  = bufA + (size_t)n * D;            // out / h_in  [n*D]
    float*    esrc  = bufB + (size_t)n * D;            // [n]
    float*    edst  = esrc + n;                        // [n]
    unsigned* mord  = (unsigned*)(edst + n);           // [n]  (denom follows)
    float*    denom = (float*)(mord + n);              // [n]
    float*    pbuf  = denom + n;                       // [etot]

    const int gemm_blocks    = (n + 15) / 16;
    const int node_blocks    = (n + 7) / 8;            // 8 waves/block
    const int edge_blocks    = (etot + 255) / 256;
    const int scatter_blocks = (etot + 7) / 8;         // wave per edge
    const int elem_blocks    = (n * D + 255) / 256;

    for (int l = 0; l < 3; ++l) {
        const float* hin = (l == 0) ? x : bufB;

        gat_gemm_wmma<<<gemm_blocks, 256, 0, stream>>>(hin, w[l], bufA, n);

        hipMemsetAsync(bufB, 0, (size_t)n * D * sizeof(float), stream);
        hipMemsetAsync(mord, 0, (size_t)n * 2 * sizeof(unsigned), stream); // mord+denom

        gat_node_logits<<<node_blocks, 256, 0, stream>>>(bufA, asrc[l], adst[l],
                                                         esrc, edst, n);
        gat_edge_logit_max<<<edge_blocks, 256, 0, stream>>>(ei, E, etot, esrc, edst,
                                                            pbuf, mord);
        gat_edge_exp_sum<<<edge_blocks, 256, 0, stream>>>(ei, E, etot, mord,
                                                          pbuf, denom);
        gat_edge_scatter<<<scatter_blocks, 256, 0, stream>>>(ei, E, etot, pbuf,
                                                             denom, bufA, bufB);
        gat_bias_relu<<<elem_blocks, 256, 0, stream>>>(bufB, bias[l], n * D);
    }

    hipMemsetAsync(d_out, 0, D * sizeof(float), stream);
    gat_colsum<<<512, 256, 0, stream>>>(bufB, (float*)d_out, n);
    gat_scale<<<1, D, 0, stream>>>((float*)d_out, 1.0f / (float)n);
}